// SparseMotionDecoder_4277787427350
// MI455X (gfx1250) — compile-verified
//
#include <hip/hip_runtime.h>
#include <hip/hip_bf16.h>

// ---------------------------------------------------------------------------
// Problem constants (from reference): T=5, N=20000, E=160000, F=64, H=4,
// HD=64. Output: thetas stacked [T, N, 6], fp32.
// ---------------------------------------------------------------------------
#define NEG_SLOPE 0.2f

typedef __attribute__((ext_vector_type(2))) float v2f;
typedef __attribute__((ext_vector_type(8))) float v8f;

// Monotonic float -> uint mapping so atomicMax on uint == float max.
__device__ __forceinline__ unsigned fkey(float x) {
  unsigned u = __float_as_uint(x);
  return (u & 0x80000000u) ? ~u : (u | 0x80000000u);
}
__device__ __forceinline__ float funkey(unsigned k) {
  unsigned u = (k & 0x80000000u) ? (k & 0x7fffffffu) : ~k;
  return __uint_as_float(u);
}
__device__ __forceinline__ float lrelu(float v) {
  return v > 0.0f ? v : NEG_SLOPE * v;
}

__device__ __forceinline__ v8f wmma_f32(v2f a, v2f b, v8f c) {
  return __builtin_amdgcn_wmma_f32_16x16x4_f32(
      /*neg_a=*/false, a, /*neg_b=*/false, b,
      /*c_mod=*/(short)0, c, /*reuse_a=*/false, /*reuse_b=*/false);
}

// ---------------------------------------------------------------------------
// Fused dual GEMM: Ya = X*Wa, Yb = X*Wb.  X:[M x 64], W:[64 x 256].
// One wave -> 16x64 strip of BOTH outputs (4 N-tiles x 2 matrices = 8 accums).
// A fragment is loaded once per k-step and reused by 8 WMMAs.
// fp32 WMMA lane layouts (16x16x4):
//   A: lane L holds row M=L%16, K = k0 + 2*(L/16) + {0,1}
//   B: lane L holds col N=L%16, K = k0 + (L/16) + {0,2}
//   C/D: lane L, vgpr v -> row v + 8*(L/16), col L%16
// ---------------------------------------------------------------------------
__global__ void wmma_dual_gemm_k64(const float* __restrict__ X,
                                   const float* __restrict__ Wa,
                                   const float* __restrict__ Wb,
                                   float* __restrict__ Ya,
                                   float* __restrict__ Yb,
                                   int M) {
  const int Nout = 256;
  int wave = (blockIdx.x * blockDim.x + threadIdx.x) >> 5;  // wave-uniform
  int lane = threadIdx.x & 31;
  const int groups = Nout / 64;            // 4 strips of 64 cols
  int total = (M >> 4) * groups;
  if (wave >= total) return;               // wave-uniform exit, EXEC stays full
  int tileM = wave / groups;
  int g = wave - tileM * groups;
  int half = lane >> 4;                    // 0 | 1
  int l16 = lane & 15;
  int colbase = g * 64 + l16;

  const float* xp  = X  + (size_t)(tileM * 16 + l16) * 64 + 2 * half;
  const float* wpa = Wa + (size_t)half * Nout + colbase;
  const float* wpb = Wb + (size_t)half * Nout + colbase;

  v8f accA[4] = {}, accB[4] = {};
#pragma unroll
  for (int k0 = 0; k0 < 64; k0 += 4) {
    v2f a;
    a.x = xp[0];
    a.y = xp[1];
#pragma unroll
    for (int j = 0; j < 4; ++j) {
      v2f b0, b1;
      b0.x = wpa[16 * j];
      b0.y = wpa[2 * Nout + 16 * j];
      b1.x = wpb[16 * j];
      b1.y = wpb[2 * Nout + 16 * j];
      accA[j] = wmma_f32(a, b0, accA[j]);
      accB[j] = wmma_f32(a, b1, accB[j]);
    }
    xp += 4;
    wpa += 4 * Nout;
    wpb += 4 * Nout;
  }

#pragma unroll
  for (int j = 0; j < 4; ++j) {
#pragma unroll
    for (int v = 0; v < 8; ++v) {
      size_t idx = (size_t)(tileM * 16 + v + 8 * half) * Nout + colbase + 16 * j;
      Ya[idx] = accA[j][v];
      Yb[idx] = accB[j][v];
    }
  }
}

// ---------------------------------------------------------------------------
// h1 GEMM: Y[M x 64] = lrelu(X[M x 64] * W[64 x 64] + bias).
// One wave computes a full 16x64 row strip (4 accumulators).
// ---------------------------------------------------------------------------
__global__ void wmma_gemm_h1(const float* __restrict__ X,
                             const float* __restrict__ W,
                             const float* __restrict__ bias,  // [64]
                             float* __restrict__ Y,
                             int M) {
  const int Nout = 64;
  int wave = (blockIdx.x * blockDim.x + threadIdx.x) >> 5;
  int lane = threadIdx.x & 31;
  if (wave >= (M >> 4)) return;            // wave-uniform exit
  int tileM = wave;
  int half = lane >> 4;
  int l16 = lane & 15;

  const float* xp = X + (size_t)(tileM * 16 + l16) * 64 + 2 * half;
  const float* wp = W + (size_t)half * Nout + l16;

  v8f acc[4] = {};
#pragma unroll
  for (int k0 = 0; k0 < 64; k0 += 4) {
    v2f a;
    a.x = xp[0];
    a.y = xp[1];
#pragma unroll
    for (int j = 0; j < 4; ++j) {
      v2f b;
      b.x = wp[16 * j];
      b.y = wp[2 * Nout + 16 * j];
      acc[j] = wmma_f32(a, b, acc[j]);
    }
    xp += 4;
    wp += 4 * Nout;
  }

#pragma unroll
  for (int j = 0; j < 4; ++j) {
    int col = l16 + 16 * j;
    float bv = bias[col];
#pragma unroll
    for (int v = 0; v < 8; ++v) {
      int orow = tileM * 16 + v + 8 * half;
      Y[(size_t)orow * Nout + col] = lrelu(acc[j][v] + bv);
    }
  }
}

// ---------------------------------------------------------------------------
// Per-step init: zero x_next [N*64], reset mkey to key(-inf) and ssum [N*H]
// ---------------------------------------------------------------------------
__global__ void init_step_kernel(float* __restrict__ xnext,
                                 unsigned* __restrict__ mkey,
                                 float* __restrict__ ssum, int N) {
  int i = blockIdx.x * blockDim.x + threadIdx.x;
  if (i < N * 64) xnext[i] = 0.0f;
  if (i < N * 4) { mkey[i] = 0x007FFFFFu; /* key(-inf) */ ssum[i] = 0.0f; }
}

// ---------------------------------------------------------------------------
// Edge pass 1: one wave per (edge, head). logits[e,h] = sum_c
//   lrelu(xl[src][h][c] + xr[tgt][h][c]) * att[h][c];  atomic segment-max.
// ---------------------------------------------------------------------------
__global__ void edge_logits_kernel(const float* __restrict__ xl,
                                   const float* __restrict__ xr,
                                   const int* __restrict__ src,
                                   const int* __restrict__ tgt,
                                   const float* __restrict__ att,   // [H*64]
                                   float* __restrict__ logits,      // [E*H]
                                   unsigned* __restrict__ mkey,     // [N*H]
                                   int E) {
  int gw = (blockIdx.x * blockDim.x + threadIdx.x) >> 5;
  int lane = threadIdx.x & 31;
  if (gw >= E * 4) return;
  int e = gw >> 2, h = gw & 3;
  int s = src[e], t = tgt[e];
  const float* pl = xl + (size_t)s * 256 + h * 64;
  const float* pr = xr + (size_t)t * 256 + h * 64;
  const float* pa = att + h * 64;

  float acc = lrelu(pl[lane] + pr[lane]) * pa[lane] +
              lrelu(pl[lane + 32] + pr[lane + 32]) * pa[lane + 32];
#pragma unroll
  for (int off = 16; off > 0; off >>= 1) acc += __shfl_xor(acc, off, 32);

  if (lane == 0) {
    logits[(size_t)e * 4 + h] = acc;
    atomicMax(mkey + (size_t)t * 4 + h, fkey(acc));
  }
}

// ---------------------------------------------------------------------------
// Edge pass 2: p = exp(logit - m[tgt]); segment-sum. p overwrites logits.
// ---------------------------------------------------------------------------
__global__ void edge_softmax_kernel(const int* __restrict__ tgt,
                                    const unsigned* __restrict__ mkey,
                                    float* __restrict__ ssum,
                                    float* __restrict__ pe,   // in: logits, out: p
                                    int E) {
  int i = blockIdx.x * blockDim.x + threadIdx.x;
  if (i >= E * 4) return;
  int e = i >> 2, h = i & 3;
  int t = tgt[e];
  float m = funkey(mkey[(size_t)t * 4 + h]);
  if (!__builtin_isfinite(m)) m = 0.0f;    // matches reference isfinite guard
  float p = __expf(pe[i] - m);
  pe[i] = p;
  atomicAdd(ssum + (size_t)t * 4 + h, p);
}

// ---------------------------------------------------------------------------
// Edge pass 3: alpha = p / (s+eps); scatter 0.25*alpha*xl[src][h][:] into the
// 64 channels of x_next[tgt] (head-mean folded into the scatter).
// One wave per (edge, head); each lane handles 2 channels.
// ---------------------------------------------------------------------------
__global__ void edge_scatter_kernel(const float* __restrict__ xl,
                                    const int* __restrict__ src,
                                    const int* __restrict__ tgt,
                                    const float* __restrict__ pe,
                                    const float* __restrict__ ssum,
                                    float* __restrict__ xnext,   // [N*64]
                                    int E) {
  int gw = (blockIdx.x * blockDim.x + threadIdx.x) >> 5;
  int lane = threadIdx.x & 31;
  if (gw >= E * 4) return;
  int e = gw >> 2, h = gw & 3;
  int s = src[e], t = tgt[e];
  float alpha = pe[(size_t)e * 4 + h] / (ssum[(size_t)t * 4 + h] + 1e-16f);
  float scale = 0.25f * alpha;             // mean over 4 heads
  const float* pl = xl + (size_t)s * 256 + h * 64;
  float* dst = xnext + (size_t)t * 64;
  atomicAdd(dst + lane,      scale * pl[lane]);
  atomicAdd(dst + lane + 32, scale * pl[lane + 32]);
}

// x_next[n][f] += bias[f]
__global__ void bias_add_kernel(float* __restrict__ x,
                                const float* __restrict__ bias, int total) {
  int i = blockIdx.x * blockDim.x + threadIdx.x;
  if (i < total) x[i] += bias[i & 63];
}

// ---------------------------------------------------------------------------
// Head: th = h1 @ W2[64x6] + b2;  out[t][n][:] = th*(1-u) + targets[n][t][:]*u
// ---------------------------------------------------------------------------
__global__ void theta_kernel(const float* __restrict__ h1,
                             const float* __restrict__ W2,   // [64*6]
                             const float* __restrict__ b2,   // [6]
                             const float* __restrict__ u_n,  // [N]
                             const float* __restrict__ targets, // [N*T*6]
                             float* __restrict__ out,        // [N*6] (slice t)
                             int t, int N, int T) {
  int n = blockIdx.x * blockDim.x + threadIdx.x;
  if (n >= N) return;
  float th[6];
#pragma unroll
  for (int j = 0; j < 6; ++j) th[j] = b2[j];
  const float* hr = h1 + (size_t)n * 64;
#pragma unroll 4
  for (int k = 0; k < 64; ++k) {
    float hk = hr[k];
#pragma unroll
    for (int j = 0; j < 6; ++j) th[j] += hk * W2[k * 6 + j];
  }
  float u = u_n[n];
  const float* tg = targets + ((size_t)n * T + t) * 6;
  float* o = out + (size_t)n * 6;
#pragma unroll
  for (int j = 0; j < 6; ++j) o[j] = th[j] * (1.0f - u) + tg[j] * u;
}

// ---------------------------------------------------------------------------
// Host-side orchestration (graph-capture safe: only kernel launches on stream)
// ---------------------------------------------------------------------------
extern "C" void kernel_launch(void* const* d_in, const int* in_sizes, int n_in,
                              void* d_out, int out_size, void* d_ws, size_t ws_size,
                              hipStream_t stream) {
  const int Tn = 5;
  const int N  = in_sizes[0] / 64;     // 20000
  const int E  = in_sizes[3] / 2;      // 160000

  const float* x_n     = (const float*)d_in[0];
  const int*   eidx    = (const int*)  d_in[3];
  const float* u_n     = (const float*)d_in[4];
  const float* targets = (const float*)d_in[6];
  const float* Wl      = (const float*)d_in[7];   // [T,64,4,64] -> [64 x 256]
  const float* Wr      = (const float*)d_in[8];
  const float* att     = (const float*)d_in[9];   // [T,4,64]
  const float* bias    = (const float*)d_in[10];  // [T,64]
  const float* W1      = (const float*)d_in[11];  // [T,64,64]
  const float* b1      = (const float*)d_in[12];  // [T,64]
  const float* W2      = (const float*)d_in[13];  // [T,64,6]
  const float* b2      = (const float*)d_in[14];  // [T,6]
  float* out = (float*)d_out;

  // Workspace carve-up (~60 MB)
  float* ws = (float*)d_ws;
  float* xl = ws;                 ws += (size_t)N * 256;
  float* xr = ws;                 ws += (size_t)N * 256;
  float* xA = ws;                 ws += (size_t)N * 64;
  float* xB = ws;                 ws += (size_t)N * 64;
  float* h1 = ws;                 ws += (size_t)N * 64;
  float* pe = ws;                 ws += (size_t)E * 4;
  unsigned* mkey = (unsigned*)ws; ws += (size_t)N * 4;
  float* ssum = ws;               ws += (size_t)N * 4;

  const int* srcI = eidx;       // edge_index[0]
  const int* tgtI = eidx + E;   // edge_index[1]

  const int BLK = 256;
  int gInit  = (N * 64 + BLK - 1) / BLK;
  int gDual  = ((N / 16) * 4 * 32 + BLK - 1) / BLK;   // wave per 16x64 strip
  int gH1    = ((N / 16) * 32 + BLK - 1) / BLK;       // wave per 16x64 row strip
  int gEdgeW = (E * 4 * 32 + BLK - 1) / BLK;          // wave per (e,h)
  int gEdgeT = (E * 4 + BLK - 1) / BLK;               // thread per (e,h)
  int gBias  = (N * 64 + BLK - 1) / BLK;
  int gNode  = (N + BLK - 1) / BLK;

  const float* xcur = x_n;
  float* xnext = xA;
  for (int t = 0; t < Tn; ++t) {
    init_step_kernel<<<gInit, BLK, 0, stream>>>(xnext, mkey, ssum, N);

    wmma_dual_gemm_k64<<<gDual, BLK, 0, stream>>>(
        xcur, Wl + (size_t)t * 64 * 256, Wr + (size_t)t * 64 * 256, xl, xr, N);

    edge_logits_kernel<<<gEdgeW, BLK, 0, stream>>>(xl, xr, srcI, tgtI,
                                                   att + (size_t)t * 256,
                                                   pe, mkey, E);
    edge_softmax_kernel<<<gEdgeT, BLK, 0, stream>>>(tgtI, mkey, ssum, pe, E);
    edge_scatter_kernel<<<gEdgeW, BLK, 0, stream>>>(xl, srcI, tgtI, pe, ssum,
                                                    xnext, E);
    bias_add_kernel<<<gBias, BLK, 0, stream>>>(xnext, bias + (size_t)t * 64,
                                               N * 64);

    wmma_gemm_h1<<<gH1, BLK, 0, stream>>>(xnext, W1 + (size_t)t * 64 * 64,
                                          b1 + (size_t)t * 64, h1, N);
    theta_kernel<<<gNode, BLK, 0, stream>>>(h1, W2 + (size_t)t * 64 * 6,
                                            b2 + (size_t)t * 6, u_n, targets,
                                            out + (size_t)t * N * 6, t, N, Tn);

    xcur  = xnext;
    xnext = (xcur == xA) ? xB : xA;
  }
}